// JointNetMetric_76390288327525
// MI455X (gfx1250) — compile-verified
//
#include <hip/hip_runtime.h>

// ---------------------------------------------------------------------------
// JointNetMetric for MI455X (gfx1250).
//   out[0] = l1_center, out[1] = l2_center, out[2] = mean chamfer
//
// One workgroup (256 thr = 8 waves) per batch; both point clouds staged in
// LDS (~67 KB << 320 KB WGP LDS). The pairwise term is folded entirely into
// V_WMMA_F32_16X16X4_F32 by augmenting K:
//     A row  = [ x0, x1, x2, 1 ]          (gt point)
//     B col  = [-2y0,-2y1,-2y2, |y|^2 ]   (pred point, -2 pre-scale is exact)
//  => c[M,N] = |y|^2 - 2 x.y
// Row mins (over pred) then need only min(rowacc, c) with |x|^2 added once at
// the end; column mins need min_v(c + |x_M|^2) which is the true distance.
// Inner loop: 1 ds_load_b64 (pipelined 1 tile ahead via a zero pad tile),
// 1 wmma, ~12 VALU, 1 unconditional ds_min_num_f32 per half (separate
// bank-offset arrays -> conflict-free, EXEC stays all-ones for WMMA).
// ---------------------------------------------------------------------------

#define NPTS    1536            // 6 * 16 * 16
#define NB      48
#define TILES   96              // NPTS / 16
#define THREADS 256
#define CPAD    (NPTS + 16)     // col-min array stride: +16 dwords de-conflicts halves

typedef __attribute__((ext_vector_type(2))) float v2f;
typedef __attribute__((ext_vector_type(8))) float v8f;

// Native LDS float atomic-min (lowers to ds_min_num_f32; no return needed).
__device__ __forceinline__ void lds_fmin(float* addr, float val) {
    __hip_atomic_fetch_min(addr, val, __ATOMIC_RELAXED, __HIP_MEMORY_SCOPE_WORKGROUP);
}

// --------------------------- tiny center-metrics ---------------------------
__global__ void center_metrics_kernel(const float* __restrict__ gt_box,
                                      const float* __restrict__ pred_center,
                                      float* __restrict__ out) {
    __shared__ float s1[64];
    __shared__ float s2[64];
    const int t = threadIdx.x;
    float a1 = 0.f, a2 = 0.f;
    for (int idx = t; idx < NB * 3; idx += 64) {
        int b = idx / 3, d = idx - 3 * b;
        float diff = gt_box[b * 18 + 9 + d] - pred_center[idx];
        a1 += fabsf(diff);
        a2 += diff * diff;
    }
    s1[t] = a1; s2[t] = a2;
    __syncthreads();
    for (int off = 32; off > 0; off >>= 1) {
        if (t < off) { s1[t] += s1[t + off]; s2[t] += s2[t + off]; }
        __syncthreads();
    }
    if (t == 0) {
        out[0] = s1[0] * (1.0f / 144.0f);
        out[1] = s2[0] * (1.0f / 144.0f);
        out[2] = 0.0f;              // chamfer accumulator (atomicAdd target)
    }
}

// ------------------------------ chamfer core -------------------------------
__global__ void __launch_bounds__(THREADS)
chamfer_kernel(const float* __restrict__ gt_box,
               const float* __restrict__ pred_pts,
               float* __restrict__ out) {
    __shared__ float gt4[NPTS][4];       // [x, y, z, 1]           (A frags)
    __shared__ float pr4[NPTS + 16][4];  // [-2x,-2y,-2z, |y|^2]   (B frags) + pad tile
    __shared__ float x2s[NPTS];          // |gt|^2
    __shared__ float colmin[2][CPAD];    // per-lane-half column mins (true d)
    __shared__ float blockAcc;

    const int b   = blockIdx.x;
    const int tid = threadIdx.x;

    // ---- box frame (redundant per thread; ~40 flops) ----
    const float* box = gt_box + b * 18 + 9;   // gt_box[b, 1, :]
    const float cx = box[0], cy = box[1], cz = box[2];
    const float sx = box[3], sy = box[4], sz = box[5];
    float rx0 = box[6], rx1 = box[7], rx2 = box[8];
    float inv = rsqrtf(rx0 * rx0 + rx1 * rx1 + rx2 * rx2 + 1e-20f);
    rx0 *= inv; rx1 *= inv; rx2 *= inv;
    // Reference's ry slice is empty (9-col box); deterministic completion:
    float tx, ty, tz;
    if (fabsf(rx2) < 0.9f) { tx = 0.f; ty = 0.f; tz = 1.f; }
    else                   { tx = 1.f; ty = 0.f; tz = 0.f; }
    float ry0 = ty * rx2 - tz * rx1;
    float ry1 = tz * rx0 - tx * rx2;
    float ry2 = tx * rx1 - ty * rx0;
    inv = rsqrtf(ry0 * ry0 + ry1 * ry1 + ry2 * ry2 + 1e-20f);
    ry0 *= inv; ry1 *= inv; ry2 *= inv;
    const float rz0 = rx1 * ry2 - rx2 * ry1;
    const float rz1 = rx2 * ry0 - rx0 * ry2;
    const float rz2 = rx0 * ry1 - rx1 * ry0;

    const float INF = __builtin_inff();

    // ---- stage both clouds (augmented-K layout) into LDS ----
    for (int n = tid; n < NPTS; n += THREADS) {
        const int f = n >> 8;              // face 0..5
        const int p = n & 255;
        const float u = (float)(p >> 4) * (1.0f / 15.0f) - 0.5f;
        const float v = (float)(p & 15) * (1.0f / 15.0f) - 0.5f;
        const float h = 0.5f;
        float c0, c1, c2;
        switch (f) {
            case 0:  c0 =  h; c1 =  u; c2 =  v; break;
            case 1:  c0 = -h; c1 =  u; c2 =  v; break;
            case 2:  c0 =  u; c1 =  h; c2 =  v; break;
            case 3:  c0 =  u; c1 = -h; c2 =  v; break;
            case 4:  c0 =  u; c1 =  v; c2 =  h; break;
            default: c0 =  u; c1 =  v; c2 = -h; break;
        }
        const float l0 = c0 * sx, l1 = c1 * sy, l2 = c2 * sz;
        const float px = cx + l0 * rx0 + l1 * ry0 + l2 * rz0;
        const float py = cy + l0 * rx1 + l1 * ry1 + l2 * rz1;
        const float pz = cz + l0 * rx2 + l1 * ry2 + l2 * rz2;
        gt4[n][0] = px; gt4[n][1] = py; gt4[n][2] = pz; gt4[n][3] = 1.0f;
        x2s[n] = px * px + py * py + pz * pz;

        const float* q = pred_pts + (size_t)b * NPTS * 3 + (size_t)n * 3;
        const float qx = q[0], qy = q[1], qz = q[2];
        pr4[n][0] = -2.0f * qx;
        pr4[n][1] = -2.0f * qy;
        pr4[n][2] = -2.0f * qz;
        pr4[n][3] = qx * qx + qy * qy + qz * qz;   // |y|^2 rides K=3 (A.w==1)

        colmin[0][n] = INF;
        colmin[1][n] = INF;
    }
    if (tid < 16) {                        // zero pad tile for branch-free prefetch
        pr4[NPTS + tid][0] = 0.f; pr4[NPTS + tid][1] = 0.f;
        pr4[NPTS + tid][2] = 0.f; pr4[NPTS + tid][3] = 0.f;
    }
    if (tid == 0) blockAcc = 0.f;
    __syncthreads();

    // ---- tiled 1536x1536 distance sweep via WMMA f32 16x16x4 ----
    const int wave = tid >> 5;
    const int lane = tid & 31;
    const int hf   = lane >> 4;            // 0: K=0..1 / M=v ; 1: K=2..3 / M=v+8
    const int lr   = lane & 15;
    float* colrow = colmin[hf];            // disjoint per half -> no exec masking

    for (int ti = wave; ti < TILES; ti += 8) {       // this wave owns rows of ti
        const int m = ti * 16 + lr;
        v2f afrag;
        afrag.x = gt4[m][hf * 2];
        afrag.y = gt4[m][hf * 2 + 1];                // hf==1 -> [z, 1]

        float x2v[8];
        #pragma unroll
        for (int v = 0; v < 8; ++v) x2v[v] = x2s[ti * 16 + hf * 8 + v];

        float rowacc[8];
        #pragma unroll
        for (int v = 0; v < 8; ++v) rowacc[v] = INF;

        // prefetch B fragment for tj = 0
        v2f bnext;
        bnext.x = pr4[lr][hf * 2];
        bnext.y = pr4[lr][hf * 2 + 1];

        for (int tj = 0; tj < TILES; ++tj) {
            const int n  = tj * 16 + lr;
            const v2f bcur = bnext;
            // pipelined prefetch (tj==95 reads the zero pad tile; discarded)
            bnext.x = pr4[n + 16][hf * 2];
            bnext.y = pr4[n + 16][hf * 2 + 1];

            v8f c = {0.f, 0.f, 0.f, 0.f, 0.f, 0.f, 0.f, 0.f};
            // c[v] = |y_n|^2 - 2 * gt[ti*16+v+8*hf] . pred[n]
            c = __builtin_amdgcn_wmma_f32_16x16x4_f32(
                    false, afrag, false, bcur, (short)0, c, false, false);

            float colm = INF;
            #pragma unroll
            for (int v = 0; v < 8; ++v) {
                rowacc[v] = fminf(rowacc[v], c[v]);        // min_N (|y|^2 - 2x.y)
                colm      = fminf(colm, c[v] + x2v[v]);    // min_M d  (true dist)
            }
            lds_fmin(&colrow[n], colm);                    // ds_min_num_f32
        }

        // reduce rowacc over N (16 lanes within the half)
        #pragma unroll
        for (int v = 0; v < 8; ++v) {
            float r = rowacc[v];
            r = fminf(r, __shfl_xor(r, 1));
            r = fminf(r, __shfl_xor(r, 2));
            r = fminf(r, __shfl_xor(r, 4));
            r = fminf(r, __shfl_xor(r, 8));
            rowacc[v] = r;
        }
        if (lr == 0) {                     // lanes 0 and 16 cover all 16 rows
            float s = 0.f;
            #pragma unroll
            for (int v = 0; v < 8; ++v) s += x2v[v] + rowacc[v];   // + |x_M|^2
            atomicAdd(&blockAcc, s);       // ds_add_f32
        }
    }
    __syncthreads();

    // ---- dist2: combine the two half-arrays (already true distances) ----
    float cs = 0.f;
    for (int n = tid; n < NPTS; n += THREADS)
        cs += fminf(colmin[0][n], colmin[1][n]);
    atomicAdd(&blockAcc, cs);
    __syncthreads();

    if (tid == 0)
        atomicAdd(&out[2], blockAcc * (1.0f / ((float)NB * (float)NPTS)));
}

// ------------------------------- launcher ----------------------------------
extern "C" void kernel_launch(void* const* d_in, const int* in_sizes, int n_in,
                              void* d_out, int out_size, void* d_ws, size_t ws_size,
                              hipStream_t stream) {
    (void)in_sizes; (void)n_in; (void)out_size; (void)d_ws; (void)ws_size;
    const float* gt_box      = (const float*)d_in[0];   // (48, 2, 9)
    const float* pred_center = (const float*)d_in[1];   // (48, 3)
    const float* pred_pts    = (const float*)d_in[2];   // (48, 1536, 3)
    float* out = (float*)d_out;                         // 3 floats

    // Writes out[0], out[1] and zeroes the out[2] accumulator.
    center_metrics_kernel<<<1, 64, 0, stream>>>(gt_box, pred_center, out);
    // One workgroup per batch; accumulates out[2].
    chamfer_kernel<<<NB, THREADS, 0, stream>>>(gt_box, pred_pts, out);
}